// ESN_38774964749017
// MI455X (gfx1250) — compile-verified
//
#include <hip/hip_runtime.h>
#include <hip/hip_bf16.h>
#include <math.h>

// ---------------------------------------------------------------------------
// ESN on MI455X (gfx1250).  fp32 WMMA (v_wmma_f32_16x16x4_f32) for the
// reservoir matvec (L2-bandwidth bound, W is L2-resident at 37.7MB) and for
// the Gram GEMM.  Persistent cooperative kernels with grid barriers carry the
// 20000 sequential recurrence steps.
// ---------------------------------------------------------------------------

typedef __attribute__((ext_vector_type(2))) float v2f;
typedef __attribute__((ext_vector_type(8))) float v8f;

#define RES       3072
#define TRAINLEN  4000
#define TESTLEN   16000
#define INITLEN   100
#define NFEAT     (RES + 2)            // 3074
#define NT        193                  // ceil(NFEAT/16)
#define NPAD      (NT * 16)            // 3088
#define KTRAIN    (TRAINLEN - INITLEN) // 3900
#define KPAD      3904                 // padded K (tail zeroed)
#define ALPHA     0.3f
#define REGC      4.84e-07f

#define NWG_STEP  192                  // workgroups in recurrence kernels
#define NTHR_STEP 128                  // 4 waves
#define KSPAN     (RES / 4)            // 768 K-columns per wave
#define NWG_CHOL  128

// ---------------------------------------------------------------------------
// Grid-wide sense-reversal barrier.  bar[0]=arrive count, bar[1]=generation.
// Zeroed via hipMemsetAsync before each cooperative kernel.
// ---------------------------------------------------------------------------
__device__ __forceinline__ void grid_barrier(unsigned* bar, unsigned nwg) {
  __syncthreads();
  if (threadIdx.x == 0) {
    __threadfence();
    unsigned gen = __hip_atomic_load(&bar[1], __ATOMIC_RELAXED, __HIP_MEMORY_SCOPE_AGENT);
    unsigned arr = __hip_atomic_fetch_add(&bar[0], 1u, __ATOMIC_ACQ_REL, __HIP_MEMORY_SCOPE_AGENT);
    if (arr == nwg - 1u) {
      __hip_atomic_store(&bar[0], 0u, __ATOMIC_RELAXED, __HIP_MEMORY_SCOPE_AGENT);
      __hip_atomic_fetch_add(&bar[1], 1u, __ATOMIC_ACQ_REL, __HIP_MEMORY_SCOPE_AGENT);
    } else {
      while (__hip_atomic_load(&bar[1], __ATOMIC_ACQUIRE, __HIP_MEMORY_SCOPE_AGENT) == gen) {
        __builtin_amdgcn_s_sleep(2);
      }
    }
  }
  __syncthreads();
}

// ---------------------------------------------------------------------------
// fp32 WMMA matvec partial: this wave accumulates W[row0:row0+16, k0:k0+768]
// times x over its K-span.  B operand = x chunk broadcast across the 16 N
// columns, so every lane's C VGPR r holds the partial dot for row r (+8 for
// the upper half-wave).  Results dropped into LDS red[64] for the 4-wave sum.
//   A 16x4 f32 layout: lanes 0-15 M=lane K={0,1}/VGPR, lanes 16-31 K={2,3}.
// ---------------------------------------------------------------------------
__device__ __forceinline__ void wmma_matvec_partial(
    const float* __restrict__ Wrow0, const float* __restrict__ x,
    int wid, int lane, float* __restrict__ red)
{
  v8f c = {0.f, 0.f, 0.f, 0.f, 0.f, 0.f, 0.f, 0.f};
  const int m  = lane & 15;
  const int kh = (lane >> 4) << 1;           // 0 for lanes 0-15, 2 for 16-31
  const float* wr = Wrow0 + (size_t)m * RES;
  const int k0 = wid * KSPAN;
#pragma unroll 4
  for (int kb = 0; kb < KSPAN; kb += 4) {
    v2f a = *(const v2f*)(wr + k0 + kb + kh); // two consecutive K of W row
    v2f b = *(const v2f*)(x  + k0 + kb + kh); // two consecutive K of x (bcast)
    c = __builtin_amdgcn_wmma_f32_16x16x4_f32(false, a, false, b,
                                              (short)0, c, false, false);
  }
  if (lane == 0) {
#pragma unroll
    for (int r = 0; r < 8; ++r) red[wid * 16 + r] = c[r];
  } else if (lane == 16) {
#pragma unroll
    for (int r = 0; r < 8; ++r) red[wid * 16 + 8 + r] = c[r];
  }
}

// ---------------------------------------------------------------------------
// Training recurrence (persistent, 192 WGs x 128 thr, one grid barrier/step).
// Writes the state columns of Xmat (feature-major, KPAD stride).
// ---------------------------------------------------------------------------
__global__ void __launch_bounds__(NTHR_STEP)
esn_train_kernel(const float* __restrict__ data, const float* __restrict__ Win,
                 const float* __restrict__ W, float* __restrict__ Xmat,
                 float* __restrict__ xbuf, unsigned* __restrict__ bar)
{
  __shared__ float red[64];
  const int wg = blockIdx.x, tid = threadIdx.x;
  const int wid = tid >> 5, lane = tid & 31;
  const int row0 = wg * 16;
  if (tid < 16) { xbuf[row0 + tid] = 0.f; xbuf[RES + row0 + tid] = 0.f; }
  grid_barrier(bar, gridDim.x);
  for (int t = 0; t < TRAINLEN; ++t) {
    const float* x  = xbuf + (t & 1) * RES;
    float*       xn = xbuf + ((t + 1) & 1) * RES;
    wmma_matvec_partial(W + (size_t)row0 * RES, x, wid, lane, red);
    __syncthreads();
    if (tid < 16) {
      const float u   = data[t];
      const float dot = red[tid] + red[16 + tid] + red[32 + tid] + red[48 + tid];
      const int row   = row0 + tid;
      const float pre = Win[2 * row] + Win[2 * row + 1] * u + dot;
      const float xv  = (1.0f - ALPHA) * x[row] + ALPHA * tanhf(pre);
      xn[row] = xv;
      if (t >= INITLEN) {
        const int tc = t - INITLEN;
        Xmat[(size_t)(2 + row) * KPAD + tc] = xv;
        if (wg == 0 && tid == 0) { Xmat[tc] = 1.f; Xmat[KPAD + tc] = u; }
      }
    }
    grid_barrier(bar, gridDim.x);
  }
}

// ---------------------------------------------------------------------------
// Gram matrix A = Xmat * Xmat^T + reg*I  (NPAD x NPAD, zero-padded rows).
// One 16x16 output tile per wave, fp32 WMMA over K = KPAD.
// ---------------------------------------------------------------------------
__global__ void __launch_bounds__(256)
gram_kernel(const float* __restrict__ Xmat, float* __restrict__ A)
{
  const int wv = blockIdx.x * 8 + (threadIdx.x >> 5);
  if (wv >= NT * NT) return;                  // wave-uniform exit
  const int ti = wv / NT, tj = wv % NT;
  const int lane = threadIdx.x & 31;
  const int m  = lane & 15;
  const int kh = (lane >> 4) << 1;
  const float* ar = Xmat + (size_t)(ti * 16 + m) * KPAD;
  const float* br = Xmat + (size_t)(tj * 16 + m) * KPAD;  // B[k,n]=Xmat[tj*16+n,k]
  v8f c = {0.f, 0.f, 0.f, 0.f, 0.f, 0.f, 0.f, 0.f};
#pragma unroll 4
  for (int k = 0; k < KPAD; k += 4) {
    v2f a = *(const v2f*)(ar + k + kh);
    v2f b = *(const v2f*)(br + k + kh);
    c = __builtin_amdgcn_wmma_f32_16x16x4_f32(false, a, false, b,
                                              (short)0, c, false, false);
  }
  const int n  = lane & 15;
  const int mo = (lane >> 4) << 3;
#pragma unroll
  for (int r = 0; r < 8; ++r) {
    const int gi = ti * 16 + r + mo, gj = tj * 16 + n;
    float v = c[r];
    if (gi == gj) v += REGC;
    A[(size_t)gi * NPAD + gj] = v;
  }
}

// B = Xmat @ Yt,  Yt[t] = data[INITLEN+1+t]
__global__ void bvec_kernel(const float* __restrict__ Xmat,
                            const float* __restrict__ data,
                            float* __restrict__ Bv)
{
  const int i = blockIdx.x * blockDim.x + threadIdx.x;
  if (i >= NFEAT) return;
  const float* xr = Xmat + (size_t)i * KPAD;
  float s = 0.f;
  for (int t = 0; t < KTRAIN; ++t) s += xr[t] * data[INITLEN + 1 + t];
  Bv[i] = s;
}

// In-place right-looking Cholesky of A (lower triangle), cooperative grid.
__global__ void __launch_bounds__(256)
chol_kernel(float* __restrict__ A, unsigned* __restrict__ bar)
{
  const int nth = gridDim.x * blockDim.x;
  const int g   = blockIdx.x * blockDim.x + threadIdx.x;
  for (int k = 0; k < NFEAT; ++k) {
    if (g == 0) A[(size_t)k * NPAD + k] = sqrtf(A[(size_t)k * NPAD + k]);
    grid_barrier(bar, gridDim.x);
    const float dk = A[(size_t)k * NPAD + k];
    for (int i = k + 1 + g; i < NFEAT; i += nth) A[(size_t)i * NPAD + k] /= dk;
    grid_barrier(bar, gridDim.x);
    for (int i = k + 1 + g; i < NFEAT; i += nth) {
      const float lik = A[(size_t)i * NPAD + k];
      float* ai = A + (size_t)i * NPAD;
      for (int j = k + 1; j <= i; ++j) ai[j] -= lik * A[(size_t)j * NPAD + k];
    }
    grid_barrier(bar, gridDim.x);
  }
}

// L y = B ; L^T z = y   (single workgroup, deterministic tree reductions)
__global__ void __launch_bounds__(256)
trisolve_kernel(const float* __restrict__ A, const float* __restrict__ Bv,
                float* __restrict__ z, float* __restrict__ y)
{
  __shared__ float red[256];
  const int tid = threadIdx.x;
  for (int i = 0; i < NFEAT; ++i) {
    float s = 0.f;
    for (int j = tid; j < i; j += 256) s += A[(size_t)i * NPAD + j] * y[j];
    red[tid] = s; __syncthreads();
    for (int o = 128; o > 0; o >>= 1) { if (tid < o) red[tid] += red[tid + o]; __syncthreads(); }
    if (tid == 0) y[i] = (Bv[i] - red[0]) / A[(size_t)i * NPAD + i];
    __syncthreads();
  }
  for (int i = NFEAT - 1; i >= 0; --i) {
    float s = 0.f;
    for (int j = i + 1 + tid; j < NFEAT; j += 256) s += A[(size_t)j * NPAD + i] * z[j];
    red[tid] = s; __syncthreads();
    for (int o = 128; o > 0; o >>= 1) { if (tid < o) red[tid] += red[tid + o]; __syncthreads(); }
    if (tid == 0) z[i] = (y[i] - red[0]) / A[(size_t)i * NPAD + i];
    __syncthreads();
  }
}

// ---------------------------------------------------------------------------
// Generative test loop: x update + y = z·[1;u;x], y feeds back as next u.
// Deterministic readout: per-WG fixed-order partials, WG0 fixed-tree reduce.
// ---------------------------------------------------------------------------
__global__ void __launch_bounds__(NTHR_STEP)
esn_test_kernel(const float* __restrict__ data, const float* __restrict__ Win,
                const float* __restrict__ W, const float* __restrict__ z,
                float* __restrict__ xbuf, float* __restrict__ Y,
                float* __restrict__ ctrl, float* __restrict__ partial,
                unsigned* __restrict__ bar)
{
  __shared__ float red[64];
  __shared__ float yp[16];
  __shared__ float rsum[NTHR_STEP];
  const int wg = blockIdx.x, tid = threadIdx.x;
  const int wid = tid >> 5, lane = tid & 31;
  const int row0 = wg * 16;
  if (wg == 0 && tid == 0) ctrl[0] = data[TRAINLEN];   // u0
  grid_barrier(bar, gridDim.x);
  for (int s = 0; s < TESTLEN; ++s) {
    const float* x  = xbuf + (s & 1) * RES;            // train left state at parity 0
    float*       xn = xbuf + ((s + 1) & 1) * RES;
    const float  u  = ctrl[0];
    wmma_matvec_partial(W + (size_t)row0 * RES, x, wid, lane, red);
    __syncthreads();
    if (tid < 16) {
      const float dot = red[tid] + red[16 + tid] + red[32 + tid] + red[48 + tid];
      const int row   = row0 + tid;
      const float pre = Win[2 * row] + Win[2 * row + 1] * u + dot;
      const float xv  = (1.0f - ALPHA) * x[row] + ALPHA * tanhf(pre);
      xn[row] = xv;
      yp[tid] = z[2 + row] * xv;
    }
    __syncthreads();
    if (tid == 0) {
      float s16 = 0.f;
#pragma unroll
      for (int r = 0; r < 16; ++r) s16 += yp[r];
      partial[wg] = s16;
    }
    grid_barrier(bar, gridDim.x);
    if (wg == 0) {
      float v = partial[tid];
      if (tid < NWG_STEP - NTHR_STEP) v += partial[NTHR_STEP + tid];
      rsum[tid] = v; __syncthreads();
      for (int o = NTHR_STEP / 2; o > 0; o >>= 1) {
        if (tid < o) rsum[tid] += rsum[tid + o];
        __syncthreads();
      }
      if (tid == 0) {
        const float yv = rsum[0] + z[0] + z[1] * u;
        Y[s] = yv;
        ctrl[0] = yv;
      }
    }
    grid_barrier(bar, gridDim.x);
  }
}

// ---------------------------------------------------------------------------
extern "C" void kernel_launch(void* const* d_in, const int* in_sizes, int n_in,
                              void* d_out, int out_size, void* d_ws, size_t ws_size,
                              hipStream_t stream)
{
  (void)in_sizes; (void)n_in; (void)out_size; (void)ws_size;
  const float* data = (const float*)d_in[0];   // (20000,1)
  const float* Win  = (const float*)d_in[1];   // (3072,2)
  const float* W    = (const float*)d_in[2];   // (3072,3072)
  // d_in[3..6] = trainLen/testLen/initLen/errorLen scalars (compile-time here)
  float* Y = (float*)d_out;                    // (16000,)

  size_t off = 0;
  auto take = [&](size_t bytes) -> void* {
    void* p = (char*)d_ws + off;
    off += (bytes + 255) & ~(size_t)255;
    return p;
  };
  float*    Xmat = (float*)take((size_t)NPAD * KPAD * sizeof(float)); // 48.2 MB
  float*    A    = (float*)take((size_t)NPAD * NPAD * sizeof(float)); // 38.1 MB
  float*    Bv   = (float*)take(NPAD * sizeof(float));
  float*    zv   = (float*)take(NPAD * sizeof(float));
  float*    yt   = (float*)take(NPAD * sizeof(float));
  float*    xbuf = (float*)take(2 * RES * sizeof(float));
  float*    part = (float*)take(NWG_STEP * sizeof(float));
  float*    ctrl = (float*)take(16 * sizeof(float));
  unsigned* bar  = (unsigned*)take(16 * sizeof(unsigned));

  hipMemsetAsync(Xmat, 0, (size_t)NPAD * KPAD * sizeof(float), stream);
  hipMemsetAsync(bar, 0, 16 * sizeof(unsigned), stream);
  esn_train_kernel<<<NWG_STEP, NTHR_STEP, 0, stream>>>(data, Win, W, Xmat, xbuf, bar);
  gram_kernel<<<(NT * NT + 7) / 8, 256, 0, stream>>>(Xmat, A);
  bvec_kernel<<<(NFEAT + 255) / 256, 256, 0, stream>>>(Xmat, data, Bv);
  hipMemsetAsync(bar, 0, 16 * sizeof(unsigned), stream);
  chol_kernel<<<NWG_CHOL, 256, 0, stream>>>(A, bar);
  trisolve_kernel<<<1, 256, 0, stream>>>(A, Bv, zv, yt);
  hipMemsetAsync(bar, 0, 16 * sizeof(unsigned), stream);
  esn_test_kernel<<<NWG_STEP, NTHR_STEP, 0, stream>>>(data, Win, W, zv, xbuf, Y,
                                                      ctrl, part, bar);
}